// DeformAttn_85461259255920
// MI455X (gfx1250) — compile-verified
//
#include <hip/hip_runtime.h>
#include <hip/hip_bf16.h>
#include <math.h>

// ---------------------------------------------------------------------------
// Problem constants (from the reference)
// ---------------------------------------------------------------------------
#define NB      8       // batch
#define LQ      2048    // queries
#define DMODEL  256
#define MH      8       // heads
#define LLV     4       // levels
#define PP      4       // points
#define DHEAD   32      // DMODEL / MH
#define STOT    3840    // sum(LENS)

__device__ __constant__ int c_lens[LLV]   = {2048, 1024, 512, 256};
__device__ __constant__ int c_starts[LLV] = {0, 2048, 3072, 3584};

typedef __attribute__((ext_vector_type(2))) float v2f;
typedef __attribute__((ext_vector_type(8))) float v8f;

// ---------------------------------------------------------------------------
// GEMM:  Y[R x C] = X[R x K] * W[C x K]^T + bias[C]
// One wave computes a 16x64 output tile (4 accumulators) with
// V_WMMA_F32_16X16X4_F32; the A fragment is loaded once per K-step and
// reused by 4 column tiles (4x less X traffic, 4 independent WMMAs in
// flight per load group).
// Requires R%16==0, C%64==0, K%8==0, and (R/16)*(C/64) % 8 == 0.
// ---------------------------------------------------------------------------
__global__ void __launch_bounds__(256)
wmma_gemm_xwt_bias(const float* __restrict__ X,
                   const float* __restrict__ W,
                   const float* __restrict__ bias,
                   float* __restrict__ Y,
                   int R, int C, int K)
{
    const int lane = threadIdx.x & 31;
    const int wave = threadIdx.x >> 5;            // 8 waves / block
    const int cgroups = C >> 6;                   // 64-wide column groups

    const int tile = blockIdx.x * 8 + wave;       // wave-uniform
    const int tr = tile / cgroups;
    const int tg = tile - tr * cgroups;
    if (tr * 16 >= R) return;                     // wave-uniform exit

    const int lm   = lane & 15;
    const int half = lane >> 4;

    const int m0  = tr << 4;
    const int cg0 = tg << 6;

    // Per-lane row pointers. Each K-step loads float2 at column k0 + 2*half.
    const v2f* __restrict__ xr  = (const v2f*)(X + (size_t)(m0 + lm) * K);
    const v2f* __restrict__ wr0 = (const v2f*)(W + (size_t)(cg0 +  0 + lm) * K);
    const v2f* __restrict__ wr1 = (const v2f*)(W + (size_t)(cg0 + 16 + lm) * K);
    const v2f* __restrict__ wr2 = (const v2f*)(W + (size_t)(cg0 + 32 + lm) * K);
    const v2f* __restrict__ wr3 = (const v2f*)(W + (size_t)(cg0 + 48 + lm) * K);

    v8f acc0 = {}, acc1 = {}, acc2 = {}, acc3 = {};
#pragma unroll 2
    for (int k0 = 0; k0 < K; k0 += 4) {
        const int pi = (k0 >> 1) + half;          // float2 index
        v2f a  = xr[pi];
        v2f b0 = wr0[pi];
        v2f b1 = wr1[pi];
        v2f b2 = wr2[pi];
        v2f b3 = wr3[pi];
        // 8 args: (neg_a, A, neg_b, B, c_mod, C, reuse_a, reuse_b)
        acc0 = __builtin_amdgcn_wmma_f32_16x16x4_f32(false, a, false, b0, (short)0, acc0, false, false);
        acc1 = __builtin_amdgcn_wmma_f32_16x16x4_f32(false, a, false, b1, (short)0, acc1, false, false);
        acc2 = __builtin_amdgcn_wmma_f32_16x16x4_f32(false, a, false, b2, (short)0, acc2, false, false);
        acc3 = __builtin_amdgcn_wmma_f32_16x16x4_f32(false, a, false, b3, (short)0, acc3, false, false);
    }

    // Column bias (same for all 8 row-VGPRs of a lane), then store.
    const float bv0 = bias[cg0 +  0 + lm];
    const float bv1 = bias[cg0 + 16 + lm];
    const float bv2 = bias[cg0 + 32 + lm];
    const float bv3 = bias[cg0 + 48 + lm];
    float* __restrict__ yp = Y + (size_t)(m0 + 8 * half) * C + cg0 + lm;
#pragma unroll
    for (int v = 0; v < 8; ++v) {
        float* row = yp + (size_t)v * C;
        row[ 0] = acc0[v] + bv0;
        row[16] = acc1[v] + bv1;
        row[32] = acc2[v] + bv2;
        row[48] = acc3[v] + bv3;
    }
}

// ---------------------------------------------------------------------------
// Fused softmax + 1D bilinear sampling + weighted accumulation.
// grid.x = N*LQ blocks, 256 threads = 8 waves; wave = head m, lane = channel d.
// All sample-point math is lane-uniform within a wave (lane only picks the
// channel), so the bounds branches are wave-uniform scalar branches.
// ---------------------------------------------------------------------------
__global__ void __launch_bounds__(256)
deform_sample_kernel(const float* __restrict__ V,      // (N, STOT, MH*DHEAD)
                     const float* __restrict__ offs,   // (N*LQ, 128)
                     const float* __restrict__ logits, // (N*LQ, 128)
                     const float* __restrict__ ref,    // (N, LQ, LLV, 1)
                     float* __restrict__ out)          // (N, LQ, 256)
{
    const int lane = threadIdx.x & 31;            // channel d within head
    const int m    = threadIdx.x >> 5;            // head
    const int nq   = blockIdx.x;                  // n*LQ + q
    const int n    = nq >> 11;                    // LQ = 2048

    // ---- softmax over the 16 (level, point) logits of this head ----
    const float* __restrict__ lg = logits + (size_t)nq * (MH * LLV * PP) + m * (LLV * PP);
    float e[LLV * PP];
    float mx = lg[0];
#pragma unroll
    for (int i = 1; i < LLV * PP; ++i) mx = fmaxf(mx, lg[i]);
    float s = 0.f;
#pragma unroll
    for (int i = 0; i < LLV * PP; ++i) { e[i] = __expf(lg[i] - mx); s += e[i]; }
    const float inv = __frcp_rn(s);

    const float* __restrict__ of = offs + (size_t)nq * (MH * LLV * PP) + m * (LLV * PP);
    const float* __restrict__ rf = ref  + (size_t)nq * LLV;
    const float* __restrict__ Vn = V + (size_t)n * STOT * DMODEL + m * DHEAD + lane;

    float acc = 0.f;
#pragma unroll
    for (int l = 0; l < LLV; ++l) {
        const int   T  = c_lens[l];
        const int   st = c_starts[l];
        const float r  = rf[l];
        const float Tf = (float)T;
#pragma unroll
        for (int p = 0; p < PP; ++p) {
            const float aw  = e[l * PP + p] * inv;
            const float loc = r + of[l * PP + p] / Tf;     // T power of 2 -> exact
            const float ix  = loc * Tf - 0.5f;
            const float fi  = floorf(ix);
            const float w1  = ix - fi;
            const int   i0  = (int)fi;
            float v0 = 0.f, v1 = 0.f;
            if (i0 >= 0 && i0 < T)
                v0 = Vn[(size_t)(st + i0) * DMODEL];
            if (i0 + 1 >= 0 && i0 + 1 < T)
                v1 = Vn[(size_t)(st + i0 + 1) * DMODEL];
            acc += aw * (v0 + (v1 - v0) * w1);
        }
    }

    out[(size_t)nq * DMODEL + m * DHEAD + lane] = acc;
}

// ---------------------------------------------------------------------------
// Host-side launch
// ---------------------------------------------------------------------------
extern "C" void kernel_launch(void* const* d_in, const int* in_sizes, int n_in,
                              void* d_out, int out_size, void* d_ws, size_t ws_size,
                              hipStream_t stream)
{
    const float* query   = (const float*)d_in[0];   // (N, LQ, 256)
    const float* refpts  = (const float*)d_in[1];   // (N, LQ, L, 1)
    const float* flat    = (const float*)d_in[2];   // (N, STOT, 256)
    // d_in[3]/d_in[4]: temporal lens / level starts (compile-time constants)
    const float* W_val   = (const float*)d_in[5];   // (256, 256)
    const float* b_val   = (const float*)d_in[6];   // (256)
    const float* W_off   = (const float*)d_in[7];   // (128, 256)
    const float* b_off   = (const float*)d_in[8];   // (128)
    const float* W_attn  = (const float*)d_in[9];   // (128, 256)
    const float* b_attn  = (const float*)d_in[10];  // (128)
    float* outp          = (float*)d_out;           // (N, LQ, 256)

    // workspace carve-up
    char* ws = (char*)d_ws;
    float* V      = (float*)ws;                                       // 31.5 MB
    float* offsB  = (float*)(ws + (size_t)NB * STOT * DMODEL * 4);    // 8 MB
    float* attnB  = (float*)((char*)offsB + (size_t)NB * LQ * 128 * 4);

    // 1) value projection: (30720 x 256) @ (256 x 256)^T + b
    {
        const int R = NB * STOT, C = DMODEL, K = DMODEL;
        const int blocks = (R / 16) * (C / 64) / 8;   // 960
        wmma_gemm_xwt_bias<<<blocks, 256, 0, stream>>>(flat, W_val, b_val, V, R, C, K);
    }
    // 2) offset projection: (16384 x 256) @ (128 x 256)^T + b
    {
        const int R = NB * LQ, C = MH * LLV * PP, K = DMODEL;
        const int blocks = (R / 16) * (C / 64) / 8;   // 256
        wmma_gemm_xwt_bias<<<blocks, 256, 0, stream>>>(query, W_off, b_off, offsB, R, C, K);
    }
    // 3) attention logits: (16384 x 256) @ (128 x 256)^T + b
    {
        const int R = NB * LQ, C = MH * LLV * PP, K = DMODEL;
        const int blocks = (R / 16) * (C / 64) / 8;   // 256
        wmma_gemm_xwt_bias<<<blocks, 256, 0, stream>>>(query, W_attn, b_attn, attnB, R, C, K);
    }
    // 4) fused softmax + bilinear sampling + head merge
    {
        deform_sample_kernel<<<NB * LQ, 256, 0, stream>>>(V, offsB, attnB, refpts, outp);
    }
}